// ScalarBorn_62156766707999
// MI455X (gfx1250) — compile-verified
//
#include <hip/hip_runtime.h>

typedef float v2f __attribute__((ext_vector_type(2)));
typedef float v8f __attribute__((ext_vector_type(8)));

namespace {
constexpr int kNZ = 250, kNY = 350;
constexpr int kFDPAD = 2, kPMLW = 20;
constexpr int kPAD = kFDPAD + kPMLW;          // 22
constexpr int kNXP = kNZ + 2 * kPAD;          // 294 valid rows
constexpr int kNYP = kNY + 2 * kPAD;          // 394 valid cols
constexpr int kShots = 4, kSrc = 1, kRecv = 100, kNT = 48;
constexpr float kDT = 0.001f, kDX = 5.0f, kDY = 5.0f;
constexpr float kFreq = 25.0f;

constexpr int kTX = (kNXP + 15) / 16;         // 19
constexpr int kTY = (kNYP + 15) / 16;         // 25
constexpr int kTiles = kTX * kTY;             // 475

// apron layout: logical (X,Y) -> memory ((X+2)*kYS + (Y+2)) within shot block.
// X writes reach 303, X reads reach 305 -> 308 rows.  Y writes reach 399 (col
// 401), Y reads reach 401 (col 403) -> stride 404.  All apron cells stay zero
// by induction (pml/zx/v2dt2 coefficients are zero outside the valid range).
constexpr int kXR = 308;                      // rows per shot block
constexpr int kYS = 404;                      // row stride (floats)
constexpr int kGS = kXR * kYS;                // per-grid coeff size
constexpr int kFS = kShots * kXR * kYS;       // per-field size (all shots)

// workspace layout (float offsets)
constexpr int OFF_VMAX = 0;
constexpr int OFF_AX = 16;
constexpr int OFF_BX = OFF_AX + 512;
constexpr int OFF_AY = OFF_BX + 512;
constexpr int OFF_BY = OFF_AY + 512;
constexpr int OFF_C1 = OFF_BY + 512;          // 2064: d1 WMMA operand table
constexpr int OFF_C2 = OFF_C1 + 320;          // d2 WMMA operand table
constexpr int OFF_V2DT2 = OFF_C2 + 320;       // 2704
constexpr int OFF_SCT = OFF_V2DT2 + kGS;
constexpr int OFF_FIELDS = OFF_SCT + kGS;
constexpr int kNumFields = 12;
constexpr size_t kTotalFloats = (size_t)OFF_FIELDS + (size_t)kNumFields * kFS;
} // namespace

// ---------------- banded FD operator operand values -------------------------
// WMMA chunk ch covers K = ch*4 .. ch*4+3.  Lane (m = lane&15, half = lane>>4)
// supplies the pair at K = ch*4 + 2*half + {0,1}.  Value depends on d = K - m:
// d1: d=0->+1, 1->-8, 3->+8, 4->-1          (x 1/(12 dx))
// d2: d=0->-1, 1->16, 2->-30, 3->16, 4->-1  (x 1/(12 dx^2))
// Per-(lane,chunk) constants -> computed ONCE per launch into ws, loaded as
// 10 b64s per wave, held in VGPRs (cannot be rematerialized).

__device__ __forceinline__ float coef_val(int d, int which) {
  float f = 0.0f;
  if (which == 0) {
    f += (d == 0) ? 1.0f : 0.0f;
    f += (d == 1) ? -8.0f : 0.0f;
    f += (d == 3) ? 8.0f : 0.0f;
    f += (d == 4) ? -1.0f : 0.0f;
  } else {
    f += (d == 0) ? -1.0f : 0.0f;
    f += (d == 1) ? 16.0f : 0.0f;
    f += (d == 2) ? -30.0f : 0.0f;
    f += (d == 3) ? 16.0f : 0.0f;
    f += (d == 4) ? -1.0f : 0.0f;
  }
  return f;
}

// acc(16x16) += D(16x20) @ tile(20x16)   -- derivative along x
__device__ __forceinline__ v8f wmma_dx5(const float* tcol, v8f acc,
                                        const v2f* A) {
#pragma unroll
  for (int ch = 0; ch < 5; ++ch) {
    v2f B;
    B.x = tcol[ch * 80];
    B.y = tcol[ch * 80 + 20];
    acc = __builtin_amdgcn_wmma_f32_16x16x4_f32(false, A[ch], false, B,
                                                (short)0, acc, false, false);
  }
  return acc;
}

// acc(16x16) += tile(16x20) @ D^T(20x16) -- derivative along y
__device__ __forceinline__ v8f wmma_dy5(const float* trow, v8f acc,
                                        const v2f* Bc) {
#pragma unroll
  for (int ch = 0; ch < 5; ++ch) {
    v2f A;
    A.x = trow[ch * 4];
    A.y = trow[ch * 4 + 1];
    acc = __builtin_amdgcn_wmma_f32_16x16x4_f32(false, A, false, Bc[ch],
                                                (short)0, acc, false, false);
  }
  return acc;
}

// load 20x20 halo tile; apron layout makes every address in-bounds and
// guarantees zero-fill semantics without per-element guards.
// f = shot block base; element (xl,yl) -> f[(X0+xl)*kYS + (Y0+yl)]
__device__ __forceinline__ void load_tile(const float* f, int X0, int Y0,
                                          float* t, int lane) {
  const float* base = f + X0 * kYS + Y0;   // logical (X0-2,Y0-2) + apron shift
#pragma unroll
  for (int it = 0; it < 12; ++it) {        // 12*32 = 384 unguarded
    int idx = lane + it * 32;
    int xl = idx / 20, yl = idx - xl * 20;
    t[idx] = base[xl * kYS + yl];
  }
  int idx = lane + 384;                    // masked tail: 384..399
  if (idx < 400) {
    int xl = idx / 20, yl = idx - xl * 20;
    t[idx] = base[xl * kYS + yl];
  }
}

// ---------------- per-step kernel 1: C-PML auxiliary px/py update -----------

__global__ __launch_bounds__(32) void pml_update_kernel(
    const float* __restrict__ wfc, const float* __restrict__ wsc,
    float* __restrict__ px, float* __restrict__ py,
    float* __restrict__ pxs, float* __restrict__ pys,
    const float* __restrict__ ax, const float* __restrict__ bx,
    const float* __restrict__ ay, const float* __restrict__ by,
    const float* __restrict__ c1g) {
  __shared__ float tile[400];
  int lane = threadIdx.x;
  int m = lane & 15, half = lane >> 4;
  int bid = blockIdx.x;
  int shot = bid / kTiles;
  int rem = bid - shot * kTiles;
  int X0 = (rem / kTY) * 16;
  int Y0 = (rem % kTY) * 16;
  int fo = shot * kXR * kYS;
  int Y = Y0 + m;

  v2f A1[5];
#pragma unroll
  for (int ch = 0; ch < 5; ++ch) A1[ch] = ((const v2f*)c1g)[ch * 32 + lane];
  const float* tcol = tile + (2 * half) * 20 + m;
  const float* trow = tile + m * 20 + 2 * half;

  for (int fld = 0; fld < 2; ++fld) {
    const float* w = (fld == 0 ? wfc : wsc) + fo;
    float* PX = (fld == 0 ? px : pxs);
    float* PY = (fld == 0 ? py : pys);
    __syncthreads();
    load_tile(w, X0, Y0, tile, lane);
    __syncthreads();
    v8f dxv = {}; v8f dyv = {};
    dxv = wmma_dx5(tcol, dxv, A1);
    dyv = wmma_dy5(trow, dyv, A1);
    union { v8f v; float f[8]; } ux, uy;
    ux.v = dxv; uy.v = dyv;
#pragma unroll
    for (int r = 0; r < 8; ++r) {
      int X = X0 + r + 8 * half;
      int g = fo + (X + 2) * kYS + (Y + 2);
      PX[g] = bx[X] * PX[g] + ax[X] * ux.f[r];
      PY[g] = by[Y] * PY[g] + ay[Y] * uy.f[r];
    }
  }
}

// ---------------- per-step kernel 2: laplacian + time update (bg + Born) ----

__global__ __launch_bounds__(32) void lap_update_kernel(
    const float* __restrict__ wfc, float* __restrict__ wfp,   // wfn in place
    const float* __restrict__ px, const float* __restrict__ py,
    float* __restrict__ zx, float* __restrict__ zy,
    const float* __restrict__ wsc, float* __restrict__ wpsc,  // wfnsc in place
    const float* __restrict__ pxs, const float* __restrict__ pys,
    float* __restrict__ zxs, float* __restrict__ zys,
    const float* __restrict__ v2dt2, const float* __restrict__ sct,
    const float* __restrict__ ax, const float* __restrict__ bx,
    const float* __restrict__ ay, const float* __restrict__ by,
    const float* __restrict__ c1g, const float* __restrict__ c2g) {
  __shared__ float tile[400];
  int lane = threadIdx.x;
  int m = lane & 15, half = lane >> 4;
  int bid = blockIdx.x;
  int shot = bid / kTiles;
  int rem = bid - shot * kTiles;
  int X0 = (rem / kTY) * 16;
  int Y0 = (rem % kTY) * 16;
  int fo = shot * kXR * kYS;
  int Y = Y0 + m;

  v2f A1[5], A2[5];
#pragma unroll
  for (int ch = 0; ch < 5; ++ch) {
    A1[ch] = ((const v2f*)c1g)[ch * 32 + lane];
    A2[ch] = ((const v2f*)c2g)[ch * 32 + lane];
  }
  const float* tcol = tile + (2 * half) * 20 + m;
  const float* trow = tile + m * 20 + 2 * half;

  float lap_s[8];

  // ---------------- background field ----------------
  v8f tx = {}; v8f ty = {};
  float wc[8];
  __syncthreads();
  load_tile(wfc + fo, X0, Y0, tile, lane);
  __syncthreads();
#pragma unroll
  for (int r = 0; r < 8; ++r) wc[r] = tile[(2 + r + 8 * half) * 20 + (2 + m)];
  tx = wmma_dx5(tcol, tx, A2);                  // d2 along x
  ty = wmma_dy5(trow, ty, A2);                  // d2 along y
  __syncthreads();
  load_tile(px + fo, X0, Y0, tile, lane);
  __syncthreads();
  tx = wmma_dx5(tcol, tx, A1);                  // + d1(px) along x
  __syncthreads();
  load_tile(py + fo, X0, Y0, tile, lane);
  __syncthreads();
  ty = wmma_dy5(trow, ty, A1);                  // + d1(py) along y
  {
    union { v8f v; float f[8]; } utx, uty;
    utx.v = tx; uty.v = ty;
#pragma unroll
    for (int r = 0; r < 8; ++r) {
      int X = X0 + r + 8 * half;
      int g = fo + (X + 2) * kYS + (Y + 2);
      int g2 = (X + 2) * kYS + (Y + 2);
      float txe = utx.f[r], tye = uty.f[r];
      float zxv = bx[X] * zx[g] + ax[X] * txe;
      float zyv = by[Y] * zy[g] + ay[Y] * tye;
      zx[g] = zxv; zy[g] = zyv;
      float lap = txe + zxv + tye + zyv;
      lap_s[r] = lap;
      wfp[g] = v2dt2[g2] * lap + 2.0f * wc[r] - wfp[g];
    }
  }

  // ---------------- scattered (Born) field ----------------
  v8f txs = {}; v8f tys = {};
  float wcs[8];
  __syncthreads();
  load_tile(wsc + fo, X0, Y0, tile, lane);
  __syncthreads();
#pragma unroll
  for (int r = 0; r < 8; ++r) wcs[r] = tile[(2 + r + 8 * half) * 20 + (2 + m)];
  txs = wmma_dx5(tcol, txs, A2);
  tys = wmma_dy5(trow, tys, A2);
  __syncthreads();
  load_tile(pxs + fo, X0, Y0, tile, lane);
  __syncthreads();
  txs = wmma_dx5(tcol, txs, A1);
  __syncthreads();
  load_tile(pys + fo, X0, Y0, tile, lane);
  __syncthreads();
  tys = wmma_dy5(trow, tys, A1);
  {
    union { v8f v; float f[8]; } utx, uty;
    utx.v = txs; uty.v = tys;
#pragma unroll
    for (int r = 0; r < 8; ++r) {
      int X = X0 + r + 8 * half;
      int g = fo + (X + 2) * kYS + (Y + 2);
      int g2 = (X + 2) * kYS + (Y + 2);
      float txe = utx.f[r], tye = uty.f[r];
      float zxv = bx[X] * zxs[g] + ax[X] * txe;
      float zyv = by[Y] * zys[g] + ay[Y] * tye;
      zxs[g] = zxv; zys[g] = zyv;
      float lapsc = txe + zxv + tye + zyv;
      wpsc[g] = v2dt2[g2] * lapsc + 2.0f * wcs[r] - wpsc[g]
              + sct[g2] * lap_s[r];
    }
  }
}

// ---------------- per-step kernel 3: source inject + receiver gather --------

__global__ void srcrec_kernel(const float* __restrict__ amp,
                              const int* __restrict__ sloc,
                              const int* __restrict__ rloc,
                              const float* __restrict__ v2dt2,
                              float* __restrict__ wfn,
                              const float* __restrict__ wfnsc,
                              float* __restrict__ out, int t) {
  int tid = blockIdx.x * blockDim.x + threadIdx.x;
  if (tid < kShots * kSrc) {
    int shot = tid / kSrc, s = tid % kSrc;
    int sx = sloc[(shot * kSrc + s) * 2 + 0] + kPAD;
    int sy = sloc[(shot * kSrc + s) * 2 + 1] + kPAD;
    int g = (shot * kXR + sx + 2) * kYS + (sy + 2);
    wfn[g] += v2dt2[(sx + 2) * kYS + (sy + 2)] * amp[(shot * kSrc + s) * kNT + t];
  }
  int rt = tid - kShots * kSrc;
  if (rt >= 0 && rt < kShots * kRecv) {
    int shot = rt / kRecv, r = rt % kRecv;
    int rx = rloc[(shot * kRecv + r) * 2 + 0] + kPAD;
    int ry = rloc[(shot * kRecv + r) * 2 + 1] + kPAD;
    out[(shot * kRecv + r) * kNT + t] =
        wfnsc[(shot * kXR + rx + 2) * kYS + (ry + 2)];
  }
}

// ---------------- setup kernels ---------------------------------------------

__global__ void vmax_kernel(const float* __restrict__ v, int n,
                            float* __restrict__ out) {
  int tid = blockIdx.x * blockDim.x + threadIdx.x;
  int stride = gridDim.x * blockDim.x;
  float mx = 0.0f;
  for (int i = tid; i < n; i += stride) mx = fmaxf(mx, fabsf(v[i]));
  atomicMax((int*)out, __float_as_int(mx));  // all |v| >= 0: int-bit monotone
}

__device__ __forceinline__ void pml_prof(int x, int n, float d, float vmax,
                                         float* a, float* b) {
  float fx = (float)x;
  float fl = fminf(fmaxf(((float)(kFDPAD + kPMLW) - fx) / (float)kPMLW, 0.f), 1.f);
  float fh = fminf(fmaxf((fx - (float)(n - 1 - kFDPAD - kPMLW)) / (float)kPMLW, 0.f), 1.f);
  float frac = fmaxf(fl, fh);
  float sigma = 3.0f * vmax * logf(1000.0f) / (2.0f * (float)kPMLW * d) * frac * frac;
  float alpha = 3.14159265358979f * kFreq * (1.0f - frac);
  float bb = expf(-(sigma + alpha) * kDT);
  *b = bb;
  *a = sigma / (sigma + alpha + 1e-9f) * (bb - 1.0f);
}

__global__ void coeff_kernel(const float* __restrict__ v,
                             const float* __restrict__ scatter,
                             float* __restrict__ ws) {
  int idx = blockIdx.x * blockDim.x + threadIdx.x;
  float vmax = ws[OFF_VMAX];
  if (idx < kNXP * kNYP) {
    int X = idx / kNYP, Y = idx - X * kNYP;
    int vi = min(max(X - kPAD, 0), kNZ - 1);
    int vj = min(max(Y - kPAD, 0), kNY - 1);
    float vp = v[vi * kNY + vj];                    // edge-pad
    int g2 = (X + 2) * kYS + (Y + 2);
    ws[OFF_V2DT2 + g2] = vp * vp * (kDT * kDT);
    float sp = 0.0f;
    int si = X - kPAD, sj = Y - kPAD;
    if (si >= 0 && si < kNZ && sj >= 0 && sj < kNY) sp = scatter[si * kNY + sj];
    ws[OFF_SCT + g2] = 2.0f * vp * sp * (kDT * kDT);
  }
  if (idx < kNXP) {        // ax/bx zero for X >= kNXP (memset) -> apron inert
    float a, b;
    pml_prof(idx, kNXP, kDX, vmax, &a, &b);
    ws[OFF_AX + idx] = a; ws[OFF_BX + idx] = b;
  }
  if (idx < kNYP) {
    float a, b;
    pml_prof(idx, kNYP, kDY, vmax, &a, &b);
    ws[OFF_AY + idx] = a; ws[OFF_BY + idx] = b;
  }
  // WMMA operand tables: [ch 0..4][lane 0..31][pair 0..1]
  if (idx < 160) {
    int ch = idx >> 5;
    int lane = idx & 31;
    int m = lane & 15, half = lane >> 4;
    const float s1 = 1.0f / (12.0f * kDX);
    const float s2 = 1.0f / (12.0f * kDX * kDX);   // dx == dy
#pragma unroll
    for (int p = 0; p < 2; ++p) {
      int d = ch * 4 + 2 * half + p - m;
      ws[OFF_C1 + ch * 64 + lane * 2 + p] = s1 * coef_val(d, 0);
      ws[OFF_C2 + ch * 64 + lane * 2 + p] = s2 * coef_val(d, 1);
    }
  }
}

// ---------------- host driver -----------------------------------------------

extern "C" void kernel_launch(void* const* d_in, const int* in_sizes, int n_in,
                              void* d_out, int out_size, void* d_ws, size_t ws_size,
                              hipStream_t stream) {
  (void)in_sizes; (void)n_in; (void)out_size; (void)ws_size;
  const float* v = (const float*)d_in[0];
  const float* scatter = (const float*)d_in[1];
  const float* amp = (const float*)d_in[2];
  const int* sloc = (const int*)d_in[3];
  const int* rloc = (const int*)d_in[4];
  float* ws = (float*)d_ws;
  float* out = (float*)d_out;

  // re-zero all persistent state every call (deterministic across replays;
  // also establishes the zero apron invariant)
  hipMemsetAsync(d_ws, 0, kTotalFloats * sizeof(float), stream);
  vmax_kernel<<<64, 256, 0, stream>>>(v, kNZ * kNY, ws + OFF_VMAX);
  int nco = kNXP * kNYP;
  coeff_kernel<<<(nco + 255) / 256, 256, 0, stream>>>(v, scatter, ws);

  float* F = ws + OFF_FIELDS;
  float* wfA = F + 0 * (size_t)kFS;  float* wfB = F + 1 * (size_t)kFS;
  float* px  = F + 2 * (size_t)kFS;  float* py  = F + 3 * (size_t)kFS;
  float* zx  = F + 4 * (size_t)kFS;  float* zy  = F + 5 * (size_t)kFS;
  float* wsA = F + 6 * (size_t)kFS;  float* wsB = F + 7 * (size_t)kFS;
  float* pxs = F + 8 * (size_t)kFS;  float* pys = F + 9 * (size_t)kFS;
  float* zxs = F + 10 * (size_t)kFS; float* zys = F + 11 * (size_t)kFS;
  const float* ax = ws + OFF_AX; const float* bx = ws + OFF_BX;
  const float* ay = ws + OFF_AY; const float* by = ws + OFF_BY;
  const float* v2 = ws + OFF_V2DT2; const float* sc = ws + OFF_SCT;
  const float* c1g = ws + OFF_C1; const float* c2g = ws + OFF_C2;

  float* wfc = wfA; float* wfp = wfB;
  float* wsc = wsA; float* wpsc = wsB;
  dim3 grid(kTiles * kShots);
  for (int t = 0; t < kNT; ++t) {
    pml_update_kernel<<<grid, 32, 0, stream>>>(wfc, wsc, px, py, pxs, pys,
                                               ax, bx, ay, by, c1g);
    lap_update_kernel<<<grid, 32, 0, stream>>>(wfc, wfp, px, py, zx, zy,
                                               wsc, wpsc, pxs, pys, zxs, zys,
                                               v2, sc, ax, bx, ay, by, c1g, c2g);
    srcrec_kernel<<<(kShots * kSrc + kShots * kRecv + 127) / 128, 128, 0, stream>>>(
        amp, sloc, rloc, v2, wfp, wpsc, out, t);
    float* tmp = wfc; wfc = wfp; wfp = tmp;       // wfn -> wfc
    tmp = wsc; wsc = wpsc; wpsc = tmp;
  }
}